// GaussianAttentionLegacy_16956530884843
// MI455X (gfx1250) — compile-verified
//
#include <hip/hip_runtime.h>
#include <hip/hip_bf16.h>

// GaussianAttentionLegacy: y[b,s,i,j] = exp(-(L1dist(x[b,i],x[b,j])^2) / (2*sigma_s^2))
// B=4, N=2048, D=3, S=8. Output 512MB f32 -> pure store-bandwidth bound on MI455X
// (~512MB / 23.3 TB/s ~= 22us floor; inputs are 96KB and live in L2).
//
// Strategy:
//  * one wave per (i0,i1) row pair; loop over 16-wide j tiles
//  * dist^2 computed once per (i,j), broadcast over 8 sigmas via
//    V_WMMA_F32_16X16X4_F32 rank-1 outer product (bit-identical to v_mul_f32:
//    one RNE multiply plus exact-zero adds)
//  * exp(x) = exp2(x * log2e) with log2e folded into the per-sigma coefficient
//  * non-temporal b32 stores (output >> 192MB L2; don't thrash the cache)

typedef __attribute__((ext_vector_type(2))) float v2f;
typedef __attribute__((ext_vector_type(8))) float v8f;

#define B_ 4
#define N_ 2048
#define S_ 8

__device__ __forceinline__ float exp2_fast(float v) {
#if defined(__HIP_DEVICE_COMPILE__) && __has_builtin(__builtin_amdgcn_exp2f)
    return __builtin_amdgcn_exp2f(v);   // v_exp_f32
#else
    return exp2f(v);                    // HIP device overload (host pass safe)
#endif
}

__global__ __launch_bounds__(256) void
gauss_l1_wmma_kernel(const float* __restrict__ x,
                     const float* __restrict__ sigmas,
                     float* __restrict__ out) {
    const int lane  = threadIdx.x & 31;
    const int wid   = threadIdx.x >> 5;
    const int b     = blockIdx.z;                 // 0..3
    const int p     = blockIdx.y * 8 + wid;       // i-pair index 0..1023
    const int i0    = 2 * p;
    const int i1    = i0 + 1;
    const int jbase = blockIdx.x * 512;           // 4 chunks of 512 columns

    const float* __restrict__ xb = x + (size_t)b * N_ * 3;

    // Wave-uniform query coordinates (i0 and i1 rows).
    const float xi00 = xb[i0 * 3 + 0], xi01 = xb[i0 * 3 + 1], xi02 = xb[i0 * 3 + 2];
    const float xi10 = xb[i1 * 3 + 0], xi11 = xb[i1 * 3 + 1], xi12 = xb[i1 * 3 + 2];

    // Per-sigma coefficient c_s = -log2(e) / (2*sigma_s^2).
    const float sig  = sigmas[lane & 7];
    const float cval = -1.44269504088896340736f / (2.0f * sig * sig);

    // A matrix 16x4 (MxK), f32, 2 VGPRs:
    //   VGPR0: lanes 0-15 -> A[M=lane][K=0], lanes 16-31 -> A[M=lane-16][K=2]
    //   VGPR1: lanes 0-15 -> A[M=lane][K=1], lanes 16-31 -> A[M=lane-16][K=3]
    // A[s][0] = c_s (rows 0-7, multiplies dist^2 of i0)
    // A[s+8][1] = c_s (rows 8-15, multiplies dist^2 of i1); everything else 0.
    v2f amat;
    amat.x = (lane < 8)                ? cval : 0.0f;
    amat.y = (lane >= 8 && lane < 16)  ? cval : 0.0f;

    const size_t plane = (size_t)N_ * N_;                // 4Mi elements per (b,s) image
    float* __restrict__ outb = out + (size_t)b * S_ * plane;

    // Fallback coefficients (same rounding behavior as the WMMA path).
    float cc[8];
#pragma unroll
    for (int r = 0; r < 8; ++r) {
        const float s = sigmas[r];
        cc[r] = -1.44269504088896340736f / (2.0f * s * s);
    }

#pragma unroll 4
    for (int t = 0; t < 32; ++t) {
        const int j = jbase + t * 16 + (lane & 15);

        // Key point coordinates for this lane's column.
        const float xj0 = xb[j * 3 + 0];
        const float xj1 = xb[j * 3 + 1];
        const float xj2 = xb[j * 3 + 2];

        const float d0 = fabsf(xi00 - xj0) + fabsf(xi01 - xj1) + fabsf(xi02 - xj2);
        const float d1 = fabsf(xi10 - xj0) + fabsf(xi11 - xj1) + fabsf(xi12 - xj2);
        const float q0 = d0 * d0;
        const float q1 = d1 * d1;

        // D-matrix ownership: lanes 0-15 -> row i0, lanes 16-31 -> row i1,
        // VGPR r -> sigma r, column j = jbase + t*16 + (lane&15).
        const int    ieff = (lane < 16) ? i0 : i1;
        float* __restrict__ o = outb + (size_t)ieff * N_ + j;

#if defined(__HIP_DEVICE_COMPILE__) && __has_builtin(__builtin_amdgcn_wmma_f32_16x16x4_f32)
        // B matrix 4x16 (KxN), f32, 2 VGPRs:
        //   VGPR0: lanes 0-15 -> B[K=0][N=lane], lanes 16-31 -> B[K=2][N=lane-16]
        //   VGPR1: lanes 0-15 -> B[K=1][N=lane], lanes 16-31 -> B[K=3][N=lane-16]
        // B[0][j] = dist^2(i0,j), B[1][j] = dist^2(i1,j); K=2,3 rows zero.
        v2f bmat;
        bmat.x = (lane < 16) ? q0 : 0.0f;
        bmat.y = (lane < 16) ? q1 : 0.0f;

        v8f acc = {};
        // 8 args: (neg_a, A, neg_b, B, c_mod, C, reuse_a, reuse_b)
        acc = __builtin_amdgcn_wmma_f32_16x16x4_f32(false, amat, false, bmat,
                                                    (short)0, acc, false, false);
#pragma unroll
        for (int r = 0; r < 8; ++r) {
            __builtin_nontemporal_store(exp2_fast(acc[r]), o + (size_t)r * plane);
        }
#else
        const float q = (lane < 16) ? q0 : q1;
#pragma unroll
        for (int r = 0; r < 8; ++r) {
            __builtin_nontemporal_store(exp2_fast(cc[r] * q), o + (size_t)r * plane);
        }
#endif
    }
}

extern "C" void kernel_launch(void* const* d_in, const int* in_sizes, int n_in,
                              void* d_out, int out_size, void* d_ws, size_t ws_size,
                              hipStream_t stream) {
    (void)in_sizes; (void)n_in; (void)out_size; (void)d_ws; (void)ws_size;
    const float* x      = (const float*)d_in[0];   // [4, 2048, 3] f32
    const float* sigmas = (const float*)d_in[1];   // [8] f32
    float*       out    = (float*)d_out;           // [4, 8, 2048, 2048] f32

    // grid: x = 4 j-chunks (512 cols), y = 128 groups of 8 i-pairs (1 per wave),
    //       z = batch. 256 threads = 8 wave32's per block.
    dim3 grid(4, N_ / 2 / 8, B_);
    dim3 block(256, 1, 1);
    gauss_l1_wmma_kernel<<<grid, block, 0, stream>>>(x, sigmas, out);
}